// Laplacian_48687749267811
// MI455X (gfx1250) — compile-verified
//
#include <hip/hip_runtime.h>
#include <stdint.h>

typedef __attribute__((ext_vector_type(2))) float v2f;
typedef __attribute__((ext_vector_type(8))) float v8f;

#define WAVES_PER_BLOCK 4
#define BLOCK_THREADS   (WAVES_PER_BLOCK * 32)
// per-wave LDS buffer: ring 16*16*3 floats (3072B) + mask 16*16 floats (1024B)
#define BUF_FLOATS      1024
#define WAVE_FLOATS     (2 * BUF_FLOATS)

__device__ __forceinline__ float sphere_sdf(float x, float y, float z) {
    float dx = x - 0.5f, dy = y - 0.5f, dz = z - 0.5f;
    // raw v_sqrt_f32: arg is a non-negative sum of squares, skip the precise-sqrt
    // denorm-scaling fixup sequence the libm sqrtf expansion emits
    return __builtin_amdgcn_sqrtf(__builtin_fmaf(dx, dx, __builtin_fmaf(dy, dy, dz * dz))) - 0.25f;
}

__device__ __forceinline__ void issue_tile_async(uint64_t ring_base, uint64_t mask_base,
                                                 unsigned ring_end, unsigned mask_end,
                                                 int tile, unsigned lds_base, int lane) {
    unsigned rb = (unsigned)tile * 3072u + (unsigned)lane * 16u;
    unsigned mb = (unsigned)tile * 1024u + (unsigned)lane * 16u;
    unsigned l  = lds_base + (unsigned)lane * 16u;
#pragma unroll
    for (int i = 0; i < 6; ++i) {           // 6 x 512B = 3KB ring tile
        unsigned go = rb + (unsigned)i * 512u;
        if (go > ring_end) go = ring_end;   // tail clamp (stores are guarded later)
        asm volatile("global_load_async_to_lds_b128 %0, %1, %2"
                     :: "v"(l + (unsigned)i * 512u), "v"(go), "s"(ring_base) : "memory");
    }
#pragma unroll
    for (int i = 0; i < 2; ++i) {           // 2 x 512B = 1KB mask tile
        unsigned go = mb + (unsigned)i * 512u;
        if (go > mask_end) go = mask_end;
        asm volatile("global_load_async_to_lds_b128 %0, %1, %2"
                     :: "v"(l + 3072u + (unsigned)i * 512u), "v"(go), "s"(mask_base) : "memory");
    }
}

__global__ __launch_bounds__(BLOCK_THREADS)
void lap_sdf_wmma_kernel(const float* __restrict__ ring,  // [N,16,3]
                         const float* __restrict__ ctr,   // [N,3]
                         const float* __restrict__ mask,  // [N,16]
                         float* __restrict__ out_center,  // [N]
                         float* __restrict__ out_lap,     // [N]
                         int Nv) {
    __shared__ float smem[WAVES_PER_BLOCK * WAVE_FLOATS];

    const int lane  = (int)(threadIdx.x & 31u);
    const int wavib = (int)(threadIdx.x >> 5);
    const int wgid  = (int)blockIdx.x * WAVES_PER_BLOCK + wavib;   // global wave id
    const int nwave = (int)gridDim.x * WAVES_PER_BLOCK;
    const int tiles = (Nv + 15) >> 4;

    float* wbase = &smem[wavib * WAVE_FLOATS];
    const unsigned lds0 = (unsigned)(uintptr_t)wbase;              // LDS byte address of buf0
    const unsigned ring_end = (unsigned)Nv * 192u - 16u;           // last legal 16B load
    const unsigned mask_end = (unsigned)Nv * 64u  - 16u;
    const uint64_t ring_u = (uint64_t)(uintptr_t)ring;
    const uint64_t mask_u = (uint64_t)(uintptr_t)mask;

    const int m  = lane & 15;          // row of A handled by this lane
    const int kh = lane >> 4;          // K-pair half within each 4-chunk

    int t = wgid;
    if (t >= tiles) return;

    issue_tile_async(ring_u, mask_u, ring_end, mask_end, t, lds0, lane);
    int buf = 0;

    for (; t < tiles; t += nwave) {
        // prefetch next tile for this wave into the other buffer (clamped: redundant
        // re-load of a valid tile at the end is harmless, keeps the wait immediate static)
        int tn = t + nwave; if (tn >= tiles) tn = tiles - 1;
        issue_tile_async(ring_u, mask_u, ring_end, mask_end, tn,
                         lds0 + (buf ? 0u : 4096u), lane);
        // 8 new loads outstanding + 8 old: <=8 means the current tile has landed
        asm volatile("s_wait_asynccnt 0x8" ::: "memory");

        // center SDF (each lane computes for its own row m; lanes l and l+16 duplicate)
        int v  = t * 16 + m;
        int vc = v < Nv ? v : (Nv - 1);
        const float* cp = ctr + 3 * (size_t)vc;
        float sc = sphere_sdf(cp[0], cp[1], cp[2]);

        const float* R = wbase + buf * BUF_FLOATS;   // ring floats in LDS
        const float* M = R + 768;                    // mask floats in LDS

        v8f acc = {};
        v2f bones; bones.x = 1.0f; bones.y = 1.0f;   // B = ones -> D rows = row-sums of A
#pragma unroll
        for (int c = 0; c < 4; ++c) {
            int k0 = 4 * c + 2 * kh;
            int p  = (m * 16 + k0) * 3;
            float s0 = sphere_sdf(R[p + 0], R[p + 1], R[p + 2]);
            float s1 = sphere_sdf(R[p + 3], R[p + 4], R[p + 5]);
            v2f a;
            a.x = M[m * 16 + k0]     * (s0 - sc);    // lap = sum_j m_j * (s_j - s_c)
            a.y = M[m * 16 + k0 + 1] * (s1 - sc);
            acc = __builtin_amdgcn_wmma_f32_16x16x4_f32(false, a, false, bones,
                                                        (short)0, acc, false, false);
        }

        // D layout: lane l holds rows {0..7}+8*(l>>4) in acc[0..7]; every column equal.
        int r = lane & 7;
        float lap = acc[0];
        lap = (r == 1) ? acc[1] : lap;
        lap = (r == 2) ? acc[2] : lap;
        lap = (r == 3) ? acc[3] : lap;
        lap = (r == 4) ? acc[4] : lap;
        lap = (r == 5) ? acc[5] : lap;
        lap = (r == 6) ? acc[6] : lap;
        lap = (r == 7) ? acc[7] : lap;

        if ((lane & 8) == 0) {                       // lanes 0-7 -> rows 0-7, 16-23 -> 8-15
            int vv = t * 16 + r + 8 * kh;
            if (vv < Nv) out_lap[vv] = lap;
        }
        if (lane < 16 && v < Nv) out_center[v] = sc;

        buf ^= 1;
    }
}

extern "C" void kernel_launch(void* const* d_in, const int* in_sizes, int n_in,
                              void* d_out, int out_size, void* d_ws, size_t ws_size,
                              hipStream_t stream) {
    (void)n_in; (void)d_ws; (void)ws_size; (void)out_size;
    const float* ring = (const float*)d_in[0];   // verts_one_ring  [N,16,3]
    const float* ctr  = (const float*)d_in[1];   // verts_central   [N,3]
    const float* mask = (const float*)d_in[2];   // one_ring_mask   [N,16]
    float* out = (float*)d_out;                  // [center_sdfs(N) | lap(N)]

    int Nv    = in_sizes[1] / 3;
    int tiles = (Nv + 15) / 16;
    int blocks = (tiles + WAVES_PER_BLOCK - 1) / WAVES_PER_BLOCK;
    if (blocks > 4096) blocks = 4096;            // ~16K waves, ~8 tiles each -> deep pipeline
    if (blocks < 1) blocks = 1;

    lap_sdf_wmma_kernel<<<blocks, BLOCK_THREADS, 0, stream>>>(
        ring, ctr, mask, out, out + Nv, Nv);
}